// reliability_based_co_teaching_loss_J_only_P1_26955214750392
// MI455X (gfx1250) — compile-verified
//
#include <hip/hip_runtime.h>
#include <math.h>

#define BB 4
#define CC 4
#define FF 32
#define NN 147456        // 384*384
#define EPSV 1e-8f

typedef __attribute__((ext_vector_type(2))) float v2f;
typedef __attribute__((ext_vector_type(8))) float v8f;

// ---------------- K0: zero the accumulator regions (ws is poisoned) ----------
__global__ void k0_zero(float* sums, float* counts, float* scal) {
    int t = threadIdx.x;
    if (t < 512) sums[t] = 0.0f;
    if (t < 16)  counts[t] = 0.0f;
    if (t < 2)   scal[t] = 0.0f;
}

// ---------------- K1: per-pixel meta (hard_aux, conf_mask), conf_aux, ce -----
__global__ void k1_meta(const float* __restrict__ pm, const float* __restrict__ pa,
                        unsigned* __restrict__ meta, float* __restrict__ confA,
                        float* __restrict__ ceArr, float* __restrict__ counts) {
    __shared__ int cnt[CC];
    int tid = threadIdx.x;
    if (tid < CC) cnt[tid] = 0;
    __syncthreads();

    long long idx = (long long)blockIdx.x * blockDim.x + tid;   // [0, B*N)
    int b = (int)(idx / NN);
    int n = (int)(idx % NN);
    const float* pmb = pm + (size_t)b * CC * NN + n;
    const float* pab = pa + (size_t)b * CC * NN + n;

    float m0 = pmb[0*NN], m1 = pmb[1*NN], m2 = pmb[2*NN], m3 = pmb[3*NN];
    float a0 = pab[0*NN], a1 = pab[1*NN], a2 = pab[2*NN], a3 = pab[3*NN];

    int am = 0; float mm = m0;                 // first-max, matches jnp.argmax
    if (m1 > mm) { mm = m1; am = 1; }
    if (m2 > mm) { mm = m2; am = 2; }
    if (m3 > mm) { mm = m3; am = 3; }
    int aa = 0; float ma = a0;
    if (a1 > ma) { ma = a1; aa = 1; }
    if (a2 > ma) { ma = a2; aa = 2; }
    if (a3 > ma) { ma = a3; aa = 3; }

    float Sm = expf(m0-mm) + expf(m1-mm) + expf(m2-mm) + expf(m3-mm);
    float Sa = expf(a0-ma) + expf(a1-ma) + expf(a2-ma) + expf(a3-ma);
    // conf_main >= conf_aux  <=>  1/Sm >= 1/Sa  <=>  Sm <= Sa
    bool msk = (Sm <= Sa);
    int label = msk ? am : aa;
    // branch-free 4-way select -> v_cndmask chain, no exec-mask branches
    float plab = ((label == 0) ? m0 : 0.0f) + ((label == 1) ? m1 : 0.0f)
               + ((label == 2) ? m2 : 0.0f) + ((label == 3) ? m3 : 0.0f);
    float ce = mm + logf(Sm) - plab;           // -log_softmax[label]

    meta[idx]  = (unsigned)aa | (msk ? 256u : 0u);
    confA[idx] = 1.0f / Sa;                    // max softmax_aux
    ceArr[idx] = ce;

    atomicAdd(&cnt[aa], 1);
    __syncthreads();
    if (tid < CC) atomicAdd(&counts[b * CC + tid], (float)cnt[tid]);
}

// ---------------- K2: segment sums of conf_aux-weighted selected features ----
// block (32,8): lane x = pixel within 32-group, wave y owns 4 channels.
// Register accumulation per (channel, class); one shuffle-reduce per wave.
__global__ void k2_sums(const float* __restrict__ fm, const float* __restrict__ fa,
                        const unsigned* __restrict__ meta, const float* __restrict__ confA,
                        float* __restrict__ sums) {
    __shared__ unsigned lmeta[256];
    __shared__ float    lw[256];
    int tx  = threadIdx.x;            // 0..31
    int ty  = threadIdx.y;            // 0..7
    int tid = ty * 32 + tx;
    int b    = blockIdx.y;
    int base = blockIdx.x * 256;      // 256-pixel window

    size_t gb = (size_t)b * NN + base;
    lmeta[tid] = meta[gb + tid];
    lw[tid]    = confA[gb + tid];
    __syncthreads();

    float acc[4][4] = {};             // [j (channel sub-index)][class]
    const float* fmb = fm + (size_t)b * FF * NN + base;
    const float* fab = fa + (size_t)b * FF * NN + base;

    for (int i = 0; i < 8; ++i) {
        int p = i * 32 + tx;          // pixel within window
        unsigned md = lmeta[p];
        int  cls = (int)(md & 0xFFu);
        bool msk = (md & 256u) != 0u;
        float w  = lw[p];
        if (i < 7) {                  // stream-ahead hint -> global_prefetch_b8
            __builtin_prefetch(&fmb[(size_t)(4*ty) * NN + p + 32], 0, 1);
            __builtin_prefetch(&fab[(size_t)(4*ty) * NN + p + 32], 0, 1);
        }
#pragma unroll
        for (int j = 0; j < 4; ++j) {
            int f = 4 * ty + j;
            size_t off = (size_t)f * NN + p;
            float vm = fmb[off];
            float va = fab[off];
            float val = w * (msk ? vm : va);
            acc[j][0] += (cls == 0) ? val : 0.0f;
            acc[j][1] += (cls == 1) ? val : 0.0f;
            acc[j][2] += (cls == 2) ? val : 0.0f;
            acc[j][3] += (cls == 3) ? val : 0.0f;
        }
    }
#pragma unroll
    for (int j = 0; j < 4; ++j) {
#pragma unroll
        for (int c = 0; c < 4; ++c) {
            float r = acc[j][c];
            for (int off = 16; off >= 1; off >>= 1) r += __shfl_xor(r, off, 32);
            if (tx == 0)
                atomicAdd(&sums[((size_t)b * CC + c) * FF + 4 * ty + j], r);
        }
    }
}

// ---------------- K3: centers = sums / max(count,1); center norms ------------
__global__ void k3_centers(const float* __restrict__ sums, const float* __restrict__ counts,
                           float* __restrict__ centers, float* __restrict__ cnorm) {
    __shared__ float sh[512];
    int tid = threadIdx.x;            // 0..511 -> (b,c,f)
    int bc  = tid >> 5;               // b*4+c
    float cv = sums[tid] / fmaxf(counts[bc], 1.0f);
    centers[tid] = cv;
    sh[tid] = cv * cv;
    __syncthreads();
    if ((tid & 31) == 0) {
        float s = 0.0f;
        for (int k = 0; k < 32; ++k) s += sh[tid + k];
        cnorm[bc] = sqrtf(s);
    }
}

// ---------------- K4: WMMA cosine similarity + fused loss reduction ----------
// Each wave: 16 pixels. A = centers (M: 4 real classes, K: 4 channels),
// B = feature chunk (N: 16 pixels). D[c][pix] = dot(center_c, feat_pix).
__global__ void k4_rel(const float* __restrict__ fm, const float* __restrict__ fa,
                       const unsigned* __restrict__ meta, const float* __restrict__ ceArr,
                       const float* __restrict__ centers, const float* __restrict__ cnorm,
                       float* __restrict__ scal /* [num, den] */) {
    __shared__ float sN[256];
    __shared__ float sD[256];
    int tid  = threadIdx.x;
    int wave = tid >> 5;
    int lane = tid & 31;
    int col  = lane & 15;             // pixel index within tile / A row
    int half = lane >> 4;             // K split: half0 -> K0,K1; half1 -> K2,K3
    int pixel = blockIdx.x * 128 + wave * 16 + col;

    float S_rel = 0.0f, S_ce = 0.0f;
    for (int b = 0; b < BB; ++b) {
        unsigned md = meta[(size_t)b * NN + pixel];
        int cls = (int)(md & 0xFFu);
        const float* fp = ((md & 256u) ? fm : fa) + (size_t)b * FF * NN + pixel;
        const float* cb = centers + (size_t)b * CC * FF;

        v8f acc = {0.f, 0.f, 0.f, 0.f, 0.f, 0.f, 0.f, 0.f};
        float n2 = 0.0f;
        int ccl = (col < CC) ? col : 0;
        float amask = (col < CC) ? 1.0f : 0.0f;   // branch-free: keep EXEC all-1s
#pragma unroll
        for (int k = 0; k < 8; ++k) {
            int fr = 4 * k + 2 * half;            // per-half K rows (B layout)
            float b0 = fp[(size_t)fr * NN];
            float b1 = fp[(size_t)(fr + 1) * NN];
            n2 = fmaf(b0, b0, n2);
            n2 = fmaf(b1, b1, n2);
            float a0 = cb[ccl * FF + fr]     * amask;
            float a1 = cb[ccl * FF + fr + 1] * amask;
            v2f A;  A.x = a0;  A.y = a1;
            v2f Bv; Bv.x = b0; Bv.y = b1;
            acc = __builtin_amdgcn_wmma_f32_16x16x4_f32(
                false, A, false, Bv, (short)0, acc, false, false);
        }
        n2 += __shfl_xor(n2, 16, 32);             // combine halves -> ||feat||^2
        // branch-free 4-way select: v_cndmask + adds, no exec-mask branching
        float dot = ((cls == 0) ? acc[0] : 0.0f) + ((cls == 1) ? acc[1] : 0.0f)
                  + ((cls == 2) ? acc[2] : 0.0f) + ((cls == 3) ? acc[3] : 0.0f);
        float rel = dot / fmaxf(sqrtf(n2) * cnorm[b * CC + cls], EPSV);
        float cev = ceArr[(size_t)b * NN + pixel];
        float keep = (half == 0) ? 1.0f : 0.0f;   // half1 is duplicate work
        S_rel += rel * keep;
        S_ce  += cev * keep;
    }
    sN[tid] = S_rel * S_ce;   // per-pixel (sum_b rel)*(sum_b ce)
    sD[tid] = S_rel;          // per-pixel sum_b rel
    __syncthreads();
    for (int s = 128; s > 0; s >>= 1) {
        if (tid < s) { sN[tid] += sN[tid + s]; sD[tid] += sD[tid + s]; }
        __syncthreads();
    }
    if (tid == 0) {
        atomicAdd(&scal[0], sN[0]);
        atomicAdd(&scal[1], sD[0]);
    }
}

// ---------------- K5: final scalar ------------------------------------------
__global__ void k5_final(const float* __restrict__ scal, float* __restrict__ out) {
    out[0] = scal[0] / scal[1];
}

extern "C" void kernel_launch(void* const* d_in, const int* in_sizes, int n_in,
                              void* d_out, int out_size, void* d_ws, size_t ws_size,
                              hipStream_t stream) {
    const float* pm = (const float*)d_in[0];
    const float* pa = (const float*)d_in[1];
    const float* fm = (const float*)d_in[2];
    const float* fa = (const float*)d_in[3];

    char* ws = (char*)d_ws;
    const size_t BN = (size_t)BB * NN;
    unsigned* meta   = (unsigned*)ws;                  // BN u32
    float*    confA  = (float*)(ws + BN * 4);          // BN f32
    float*    ceA    = (float*)(ws + BN * 8);          // BN f32
    float*    sums   = (float*)(ws + BN * 12);         // 512 f32
    float*    counts = sums + 512;                     // 16 f32
    float*    centers= counts + 16;                    // 512 f32
    float*    cnorm  = centers + 512;                  // 16 f32
    float*    scal   = cnorm + 16;                     // 2 f32 (num, den)

    k0_zero<<<1, 512, 0, stream>>>(sums, counts, scal);
    k1_meta<<<(unsigned)(BN / 256), 256, 0, stream>>>(pm, pa, meta, confA, ceA, counts);
    dim3 g2(NN / 256, BB), b2(32, 8);
    k2_sums<<<g2, b2, 0, stream>>>(fm, fa, meta, confA, sums);
    k3_centers<<<1, 512, 0, stream>>>(sums, counts, centers, cnorm);
    k4_rel<<<NN / 128, 256, 0, stream>>>(fm, fa, meta, ceA, centers, cnorm, scal);
    k5_final<<<1, 1, 0, stream>>>(scal, (float*)d_out);
}